// MIMOGeneralDotProductAttention_22256520528800
// MI455X (gfx1250) — compile-verified
//
#include <hip/hip_runtime.h>
#include <hip/hip_bf16.h>

// Problem constants (from reference)
#define BB   8
#define AA   6      // agents = queries = keys dim
#define QS   32
#define KS   1024
#define CC   256
#define HH   64
#define WW   64
#define CHW  (CC * HH * WW)          // 1048576 floats per (b, k) plane
#define N4   (CHW / 4)               // 262144 float4 per plane
#define OUT_ELEMS  ((long long)BB * AA * CHW)   // 50331648
#define ROWS (BB * AA)               // 48 rows in projection GEMM

typedef __attribute__((ext_vector_type(2))) float v2f;
typedef __attribute__((ext_vector_type(8))) float v8f;
typedef __attribute__((ext_vector_type(4))) float f4;

// ---------------------------------------------------------------------------
// Stage 1: query = qu @ W.T + b   (48 x 1024 output, K=32), via
// V_WMMA_F32_16X16X4_F32. One wave (32 lanes) per 16x16 output tile,
// 8 WMMA steps over K. Grid: (KS/16 = 64 N-tiles, 3 M-tiles), block = 32.
//
// Layouts (ISA 7.12.2, f32):
//   A 16x4 : lanes 0-15 -> {K0,K1} of row M=lane; lanes 16-31 -> {K2,K3}
//   B 4x16 : VGPR0 = rows K0 (lanes 0-15) / K2 (lanes 16-31), VGPR1 = K1/K3
//   C/D    : VGPR v = row (m0 + v + 8*half), col n0 + (lane&15)
// ---------------------------------------------------------------------------
__global__ void proj_wmma_kernel(const float* __restrict__ qu,
                                 const float* __restrict__ Wm,
                                 const float* __restrict__ bias,
                                 float* __restrict__ query) {
    const int lane   = threadIdx.x;          // 0..31, full wave, no divergence
    const int half   = lane >> 4;            // 0 or 1
    const int lane16 = lane & 15;
    const int n0     = blockIdx.x * 16;      // column tile in [0,1024)
    const int m0     = blockIdx.y * 16;      // row tile in {0,16,32}

    const int arow = m0 + lane16;            // A row for this lane (< 48)
    const int ncol = n0 + lane16;            // B/D column for this lane

    // C accumulator initialized with the bias (depends only on column n)
    const float bn = bias[ncol];
    v8f c;
#pragma unroll
    for (int i = 0; i < 8; ++i) c[i] = bn;

#pragma unroll
    for (int k0 = 0; k0 < QS; k0 += 4) {
        const int ka = k0 + half * 2;        // this lane's K pair
        v2f a, b;
        // A = qu (row-major [48 x 32])
        a.x = qu[arow * QS + ka + 0];
        a.y = qu[arow * QS + ka + 1];
        // B[k][n] = W[n][k]  (W row-major [1024 x 32])
        b.x = Wm[ncol * QS + ka + 0];
        b.y = Wm[ncol * QS + ka + 1];
        c = __builtin_amdgcn_wmma_f32_16x16x4_f32(
                /*neg_a=*/false, a, /*neg_b=*/false, b,
                /*c_mod=*/(short)0, c, /*reuse_a=*/false, /*reuse_b=*/false);
    }

    // Scatter D: VGPR v holds row m0 + v + 8*half, column ncol
#pragma unroll
    for (int i = 0; i < 8; ++i) {
        const int row = m0 + i + 8 * half;
        query[row * KS + ncol] = c[i];
    }
}

// ---------------------------------------------------------------------------
// Stage 2: attn[b][kk][q] = dot(k[b,kk,:], query[b,q,:]) over 1024.
// One wave per (b, kk, q); grid = (36, 8), block = 32.
// ---------------------------------------------------------------------------
__global__ void attn_kernel(const float* __restrict__ kmat,
                            const float* __restrict__ query,
                            float* __restrict__ attn) {
    const int pair = blockIdx.x;             // 0..35
    const int b    = blockIdx.y;             // 0..7
    const int kk   = pair / AA;
    const int q    = pair % AA;

    const float* kp = kmat  + (long long)(b * AA + kk) * KS;
    const float* qp = query + (long long)(b * AA + q)  * KS;

    float s = 0.f;
#pragma unroll 4
    for (int d = threadIdx.x; d < KS; d += 32) s += kp[d] * qp[d];

#pragma unroll
    for (int off = 16; off >= 1; off >>= 1) s += __shfl_xor(s, off, 32);

    if (threadIdx.x == 0) attn[b * (AA * AA) + kk * AA + q] = s;
}

// ---------------------------------------------------------------------------
// Stage 3: softmax over the key dim (axis kk) for each (b, q).
// 48 independent 6-element softmaxes. Writes both the scratch copy (for
// stage 4) and the attn_sm portion of d_out.
// ---------------------------------------------------------------------------
__global__ void softmax_kernel(const float* __restrict__ attn,
                               float* __restrict__ attn_sm,
                               float* __restrict__ out_tail) {
    const int t = threadIdx.x;
    if (t >= BB * AA) return;
    const int b = t / AA;
    const int q = t % AA;

    float vals[AA];
    float m = -3.402823466e+38f;
#pragma unroll
    for (int kk = 0; kk < AA; ++kk) {
        vals[kk] = attn[b * (AA * AA) + kk * AA + q];
        m = fmaxf(m, vals[kk]);
    }
    float sum = 0.f;
#pragma unroll
    for (int kk = 0; kk < AA; ++kk) {
        vals[kk] = expf(vals[kk] - m);
        sum += vals[kk];
    }
    const float inv = 1.f / sum;
#pragma unroll
    for (int kk = 0; kk < AA; ++kk) {
        const float r = vals[kk] * inv;
        const int idx = b * (AA * AA) + kk * AA + q;
        attn_sm[idx]  = r;
        out_tail[idx] = r;
    }
}

// ---------------------------------------------------------------------------
// Stage 4 (memory-bound hot loop): out[b,q,chw] = sum_k attn_sm[b,k,q] * v[b,k,chw]
// Each thread owns one float4 spatial slot: 6 NT loads (one per k-plane),
// 36 FMA4, 6 NT stores -> v read exactly once, out written exactly once.
// Grid = (N4/256 = 1024, B), block = 256 (8 wave32s).
// ---------------------------------------------------------------------------
__global__ void combine_kernel(const float* __restrict__ v,
                               const float* __restrict__ attn_sm,
                               float* __restrict__ out) {
    const int b = blockIdx.y;
    const long long i4 = (long long)blockIdx.x * blockDim.x + threadIdx.x; // < N4

    // Per-batch 6x6 attention weights (uniform across the block)
    float wgt[AA][AA];
    const float* ap = attn_sm + b * (AA * AA);
#pragma unroll
    for (int kk = 0; kk < AA; ++kk)
#pragma unroll
        for (int q = 0; q < AA; ++q) wgt[kk][q] = ap[kk * AA + q];

    const f4* vp = reinterpret_cast<const f4*>(v)   + (long long)b * AA * N4 + i4;
    f4*       op = reinterpret_cast<f4*>(out)       + (long long)b * AA * N4 + i4;

    f4 acc[AA];
#pragma unroll
    for (int q = 0; q < AA; ++q) acc[q] = (f4){0.f, 0.f, 0.f, 0.f};

#pragma unroll
    for (int kk = 0; kk < AA; ++kk) {
        const f4 x = __builtin_nontemporal_load(vp + (long long)kk * N4);
#pragma unroll
        for (int q = 0; q < AA; ++q) {
            acc[q].x += wgt[kk][q] * x.x;
            acc[q].y += wgt[kk][q] * x.y;
            acc[q].z += wgt[kk][q] * x.z;
            acc[q].w += wgt[kk][q] * x.w;
        }
    }
#pragma unroll
    for (int q = 0; q < AA; ++q)
        __builtin_nontemporal_store(acc[q], op + (long long)q * N4);
}

// ---------------------------------------------------------------------------
extern "C" void kernel_launch(void* const* d_in, const int* in_sizes, int n_in,
                              void* d_out, int out_size, void* d_ws, size_t ws_size,
                              hipStream_t stream) {
    const float* qu   = (const float*)d_in[0];   // [8, 6, 32]
    const float* kmat = (const float*)d_in[1];   // [8, 6, 1024]
    const float* v    = (const float*)d_in[2];   // [8, 6, 256, 64, 64]
    const float* Wm   = (const float*)d_in[3];   // [1024, 32]
    const float* bias = (const float*)d_in[4];   // [1024]

    float* out = (float*)d_out;                  // out (50331648) ++ attn_sm (288)
    float* ws  = (float*)d_ws;

    float* ws_query  = ws;                       // 48 * 1024 floats
    float* ws_attn   = ws_query + ROWS * KS;     // 288 floats
    float* ws_attnsm = ws_attn + BB * AA * AA;   // 288 floats

    // 1) query projection via f32 WMMA (exact f32 math)
    proj_wmma_kernel<<<dim3(KS / 16, ROWS / 16), 32, 0, stream>>>(qu, Wm, bias, ws_query);

    // 2) attention scores
    attn_kernel<<<dim3(AA * AA, BB), 32, 0, stream>>>(kmat, ws_query, ws_attn);

    // 3) softmax over key dim; also emits attn_sm output
    softmax_kernel<<<1, 64, 0, stream>>>(ws_attn, ws_attnsm, out + OUT_ELEMS);

    // 4) streaming weighted combine (the ~400 MB HBM-bound stage)
    combine_kernel<<<dim3(N4 / 256, BB), 256, 0, stream>>>(v, ws_attnsm, out);
}